// Modified_Fourier_Net_91242285236682
// MI455X (gfx1250) — compile-verified
//
#include <hip/hip_runtime.h>

// Modified Fourier Net inference, fp32 WMMA (V_WMMA_F32_16X16X4_F32) on gfx1250.
//
// Layouts (CDNA5 ISA, wave32):
//  A (16x4 f32):  VGPR0 = {A[m][0] lanes 0-15, A[m][2] lanes 16-31}, VGPR1 = {A[m][1], A[m][3]}
//  B (4x16 f32):  VGPR0 = {B[0][n] lanes 0-15, B[2][n] lanes 16-31}, VGPR1 = {B[1][n], B[3][n]}
//  C/D (16x16):   VGPR g: lanes 0-15 -> [M=g][N=lane], lanes 16-31 -> [M=g+8][N=lane-16]
// A per-lane b64 load of W[row=16j+l16][4c + 2*half] IS a B operand; a per-lane b64
// load of ACT[l16][4c + 2*half] IS an A operand.
//
// Bias trick: hidden/head biases live in weight column k=50 (the pad column) and
// activation column 50 is kept == 1.0 through every layer:
//   fourier pad -> U[50]=sin(0)=0, V[50]=cos(0)=1
//   W[i][50][50] = pi/2 -> Z[50]=sin(pi/2)=1 -> out[50] = V[50] = 1
// so hidden/head WMMA accumulators start from inline-constant 0 (no splat movs).

#define WIDTH   50
#define PW      64      // neuron dim padded to 4 blocks of 16
#define PK      52      // K padded to 13 chunks of 4 (col 50 = bias, col 51 = 0)
#define NBLK    4
#define NCHUNK  13
#define NHID    9
#define WAVES   8
#define TPW     4       // tiles per wave
#define ACT_PITCH 68    // row pitch (floats) of per-wave act tile: bank-conflict-free
#define PI_2    1.57079632679489662f

typedef float v2f __attribute__((ext_vector_type(2)));
typedef float v8f __attribute__((ext_vector_type(8)));

static __device__ __forceinline__ v8f wmma4(v2f a, v2f b, v8f c) {
    // D = A(16x4) * B(4x16) + C, fp32
    return __builtin_amdgcn_wmma_f32_16x16x4_f32(false, a, false, b, (short)0, c,
                                                 false, false);
}

static __device__ __forceinline__ v8f splat8(float v) {
    v8f r = {v, v, v, v, v, v, v, v};
    return r;
}

__global__ __launch_bounds__(256) void mfn_kernel(
    const float* __restrict__ x,
    const float* __restrict__ fw,   const float* __restrict__ fb,
    const float* __restrict__ h1w,  const float* __restrict__ h1b,
    const float* __restrict__ hw,   const float* __restrict__ hb,
    const float* __restrict__ headw,const float* __restrict__ headb,
    float* __restrict__ out, int n)
{
    __shared__ float sHW[NHID * PW * PK];   // hidden weights + bias col + pi/2 keeper
    __shared__ float sFW[PW * 4];           // fourier weights
    __shared__ float sH1W[PW * 4];
    __shared__ float sHEADW[16 * PK];       // head weights + bias col
    __shared__ float sFB[PW];
    __shared__ float sH1B[PW];
    __shared__ float sACT[WAVES * 16 * ACT_PITCH];  // per-wave private 16 x 64 act tile

    const int tid = threadIdx.x;

    // ---- cooperative weight staging (once per block, amortized over 32 tiles) ----
    for (int idx = tid; idx < NHID * PW * PK; idx += 256) {
        int i  = idx / (PW * PK);
        int rk = idx - i * (PW * PK);
        int r  = rk / PK, k = rk - r * PK;
        float v = 0.0f;
        if (r < WIDTH) {
            if (k < WIDTH)       v = hw[(i * WIDTH + r) * WIDTH + k];
            else if (k == WIDTH) v = hb[i * WIDTH + r];        // bias column
        } else if (r == WIDTH && k == WIDTH) {
            v = PI_2;                                          // keeps act[50] == 1
        }
        sHW[idx] = v;
    }
    for (int idx = tid; idx < PW * 4; idx += 256) {
        int r = idx >> 2, k = idx & 3;
        sFW[idx]  = (r < WIDTH) ? fw[r * 4 + k]  : 0.0f;
        sH1W[idx] = (r < WIDTH) ? h1w[r * 4 + k] : 0.0f;
    }
    for (int idx = tid; idx < 16 * PK; idx += 256) {
        int r = idx / PK, k = idx - r * PK;
        float v = 0.0f;
        if (r < 4) {
            if (k < WIDTH)       v = headw[r * WIDTH + k];
            else if (k == WIDTH) v = headb[r];                 // bias column
        }
        sHEADW[idx] = v;
    }
    for (int idx = tid; idx < PW; idx += 256) {
        sFB[idx]  = (idx < WIDTH) ? fb[idx] : 0.0f;
        float hv = 0.0f;
        if (idx < WIDTH)       hv = h1b[idx];
        else if (idx == WIDTH) hv = PI_2;                      // sin -> act[50] = 1
        sH1B[idx] = hv;
    }
    __syncthreads();

    const int lane = tid & 31;
    const int wave = tid >> 5;
    const int half = lane >> 4;   // 0: lanes 0-15, 1: lanes 16-31
    const int l16  = lane & 15;
    float* act = &sACT[wave * 16 * ACT_PITCH];

    const int ntiles = (n + 15) >> 4;
    const int tile0  = blockIdx.x * (WAVES * TPW) + wave * TPW;

    for (int t = 0; t < TPW; ++t) {
        const int tile = tile0 + t;
        if (tile >= ntiles) break;
        const int p0 = tile << 4;

        // ---- load x tile as a ready A operand (16x4), coalesced b64 ----
        int row = p0 + l16;
        if (row >= n) row = n - 1;
        const v2f ax = *(const v2f*)&x[row * 4 + half * 2];

        // ---- Fourier features: F = x@FW^T + fb ; U=sin(F), VmU=cos(F)-sin(F) ----
        v8f U[NBLK], VmU[NBLK];
#pragma unroll
        for (int j = 0; j < NBLK; ++j) {
            v8f c = splat8(sFB[j * 16 + l16]);              // bias as C init
            v2f b = *(const v2f*)&sFW[(j * 16 + l16) * 4 + half * 2];
            c = wmma4(ax, b, c);
            v8f u, vmu;
#pragma unroll
            for (int g = 0; g < 8; ++g) {
                float s = __sinf(c[g]);
                u[g]   = s;
                vmu[g] = __cosf(c[g]) - s;
            }
            U[j] = u;
            VmU[j] = vmu;
        }

        // ---- first layer: out = sin(x@H1W^T + b), store to act tile (D->A transpose) ----
#pragma unroll
        for (int j = 0; j < NBLK; ++j) {
            v8f c = splat8(sH1B[j * 16 + l16]);
            v2f b = *(const v2f*)&sH1W[(j * 16 + l16) * 4 + half * 2];
            c = wmma4(ax, b, c);
#pragma unroll
            for (int g = 0; g < 8; ++g)
                act[(g + half * 8) * ACT_PITCH + j * 16 + l16] = __sinf(c[g]);
        }

        // ---- 9 hidden layers: Z = sin(act@W^T + b); out = U + Z*(V-U) ----
        for (int i = 0; i < NHID; ++i) {
            // read full A row (13 chunks) into registers before overwriting act
            v2f a[NCHUNK];
#pragma unroll
            for (int cc = 0; cc < NCHUNK; ++cc)
                a[cc] = *(const v2f*)&act[l16 * ACT_PITCH + cc * 4 + half * 2];

            const float* wl = &sHW[i * PW * PK];
            // 4 independent accumulation chains, interleaved per chunk
            v8f c[NBLK];
#pragma unroll
            for (int j = 0; j < NBLK; ++j) c[j] = splat8(0.0f) * 0.0f, c[j] = (v8f){};
#pragma unroll
            for (int cc = 0; cc < NCHUNK; ++cc) {
#pragma unroll
                for (int j = 0; j < NBLK; ++j) {
                    v2f b = *(const v2f*)&wl[(j * 16 + l16) * PK + cc * 4 + half * 2];
                    c[j] = wmma4(a[cc], b, c[j]);
                }
            }
            // elementwise modulation (same C/D layout as U/VmU), store for next layer
#pragma unroll
            for (int j = 0; j < NBLK; ++j) {
#pragma unroll
                for (int g = 0; g < 8; ++g) {
                    float z = __sinf(c[j][g]);
                    act[(g + half * 8) * ACT_PITCH + j * 16 + l16] =
                        U[j][g] + z * VmU[j][g];
                }
            }
        }

        // ---- head: out16 = act@HEADW^T + hb (bias in col 50; cols 0..3 valid) ----
        {
            v2f a[NCHUNK];
#pragma unroll
            for (int cc = 0; cc < NCHUNK; ++cc)
                a[cc] = *(const v2f*)&act[l16 * ACT_PITCH + cc * 4 + half * 2];
            const float* wrow = &sHEADW[l16 * PK + half * 2];
            v8f c0 = {}, c1 = {};                    // two chains to cut latency
#pragma unroll
            for (int cc = 0; cc < NCHUNK; ++cc) {
                v2f b = *(const v2f*)&wrow[cc * 4];
                if (cc & 1) c1 = wmma4(a[cc], b, c1);
                else        c0 = wmma4(a[cc], b, c0);
            }
            v8f c = c0 + c1;
            // D -> row-major via act tile, then coalesced b64 global store
#pragma unroll
            for (int g = 0; g < 8; ++g)
                act[(g + half * 8) * ACT_PITCH + l16] = c[g];
            const int prow = p0 + l16;
            v2f r = *(const v2f*)&act[l16 * ACT_PITCH + half * 2];
            if (prow < n)
                *(v2f*)&out[prow * 4 + half * 2] = r;
        }
    }
}

extern "C" void kernel_launch(void* const* d_in, const int* in_sizes, int n_in,
                              void* d_out, int out_size, void* d_ws, size_t ws_size,
                              hipStream_t stream) {
    const float* x     = (const float*)d_in[0];
    const float* fw    = (const float*)d_in[1];
    const float* fb    = (const float*)d_in[2];
    const float* h1w   = (const float*)d_in[3];
    const float* h1b   = (const float*)d_in[4];
    const float* hw    = (const float*)d_in[5];
    const float* hb    = (const float*)d_in[6];
    const float* headw = (const float*)d_in[7];
    const float* headb = (const float*)d_in[8];

    const int n = in_sizes[0] / 4;                 // N points (x is [N,4])
    const int ntiles = (n + 15) / 16;
    const int blocks = (ntiles + WAVES * TPW - 1) / (WAVES * TPW);

    mfn_kernel<<<blocks, 256, 0, stream>>>(x, fw, fb, h1w, h1b, hw, hb,
                                           headw, headb, (float*)d_out, n);
}